// PMA_3676492005784
// MI455X (gfx1250) — compile-verified
//
#include <hip/hip_runtime.h>
#include <hip/hip_bf16.h>

// ---------------------------------------------------------------------------
// Graph hop propagation: out[k+1] = l2norm( A * out[k] + 0.1*noise ), 8 hops.
// N=100000 rows, D=128 features, E=3.2M edges. Slab (51.2MB) is L2-resident
// on MI455X (192MB L2), so the hop kernels are an L2 gather problem.
// Strategy: build CSR once (no output atomics in hops), one wave32 per row,
// fused gather+noise+normalize with a 4-deep gather pipeline (4 outstanding
// global_load_b128 per wave), WMMA f32 16x16x4 for the norm reduction,
// readlane-based column broadcast (SGPR gather base), rsqrt epilogue.
// ---------------------------------------------------------------------------

#define HOP_D 128
#define NUM_HOPS 8

typedef __attribute__((ext_vector_type(2))) float v2f;
typedef __attribute__((ext_vector_type(8))) float v8f;

// --------------------------- RNG (deterministic) ---------------------------
__device__ __forceinline__ unsigned pcg_hash(unsigned v) {
    v = v * 747796405u + 2891336453u;
    unsigned w = ((v >> ((v >> 28u) + 4u)) ^ v) * 277803737u;
    return (w >> 22u) ^ w;
}

// 4 gaussians per call via 2x Box-Muller, pure function of seed.
__device__ __forceinline__ void gauss4(unsigned seed, float g[4]) {
    const float k32 = 2.3283064365386963e-10f; // 2^-32
    unsigned r0 = pcg_hash(seed);
    unsigned r1 = pcg_hash(seed ^ 0x9E3779B9u);
    unsigned r2 = pcg_hash(seed + 0x7F4A7C15u);
    unsigned r3 = pcg_hash(seed ^ 0x94D049BBu);
    float u1 = fmaxf((float)r0 * k32, 1e-12f);
    float u2 = (float)r1 * k32;
    float u3 = fmaxf((float)r2 * k32, 1e-12f);
    float u4 = (float)r3 * k32;
    float m1 = sqrtf(-2.0f * __logf(u1));
    float m2 = sqrtf(-2.0f * __logf(u3));
    float s1, c1, s2, c2;
    __sincosf(6.28318530717958647f * u2, &s1, &c1);
    __sincosf(6.28318530717958647f * u4, &s2, &c2);
    g[0] = m1 * c1; g[1] = m1 * s1; g[2] = m2 * c2; g[3] = m2 * s2;
}

// ------------------ wave32 reduction via V_WMMA_F32_16X16X4_F32 ------------
// A (16x4 f32): lane L<16 holds A[L][0],A[L][1]; lane L>=16 holds A[L-16][2..3].
// With a = {s_lane, 0} and B = all-ones, D[m][j] = s_m + s_{m+16} for all j.
// C/D layout: c[v] on lane j<16 = D[v][j]; on lane j>=16 = D[v+8][j-16].
// Sum c[0..7] locally, then one ds_swizzle SWAPX16 fold => full 32-lane sum.
// EXEC must be all ones here (callers use wave-uniform guards only).
__device__ __forceinline__ float wave_sum_wmma(float s) {
    v2f a; a[0] = s;    a[1] = 0.0f;
    v2f b; b[0] = 1.0f; b[1] = 1.0f;
    v8f c = {};
    c = __builtin_amdgcn_wmma_f32_16x16x4_f32(
        /*neg_a=*/false, a, /*neg_b=*/false, b,
        /*c_mod=*/(short)0, c, /*reuse_a=*/false, /*reuse_b=*/false);
    float t = ((c[0] + c[1]) + (c[2] + c[3])) + ((c[4] + c[5]) + (c[6] + c[7]));
    // SWAPX16: offset[15]=0, and_mask=0x1f, or_mask=0, xor_mask=0x10 -> 0x401F
    int other = __builtin_amdgcn_ds_swizzle(__float_as_int(t), 0x401F);
    return t + __int_as_float(other);
}

// ------------------------------ CSR build ----------------------------------
__global__ void degree_kernel(const int* __restrict__ rows,
                              int* __restrict__ deg, int e) {
    int i = blockIdx.x * blockDim.x + threadIdx.x;
    if (i < e) atomicAdd(&deg[rows[i]], 1);
}

// Single-block chunked Hillis-Steele exclusive scan over N (~100 chunks).
__global__ void scan_kernel(const int* __restrict__ deg,
                            int* __restrict__ row_ptr,
                            int* __restrict__ cursor, int n) {
    __shared__ int sh[1024];
    __shared__ int carry_sh;
    const int tid = threadIdx.x;
    if (tid == 0) { carry_sh = 0; row_ptr[0] = 0; }
    __syncthreads();
    for (int base = 0; base < n; base += 1024) {
        const int i = base + tid;
        const int v = (i < n) ? deg[i] : 0;
        sh[tid] = v;
        __syncthreads();
        for (int off = 1; off < 1024; off <<= 1) {
            int t = (tid >= off) ? sh[tid - off] : 0;
            __syncthreads();
            sh[tid] += t;
            __syncthreads();
        }
        const int carry = carry_sh;
        const int inc = sh[tid] + carry;   // inclusive prefix incl. carry
        if (i < n) {
            row_ptr[i + 1] = inc;
            cursor[i] = inc - v;           // exclusive prefix
        }
        __syncthreads();
        if (tid == 1023) carry_sh = inc;
        __syncthreads();
    }
}

__global__ void fill_kernel(const int* __restrict__ rows,
                            const int* __restrict__ cols,
                            int* __restrict__ cursor,
                            int* __restrict__ csr_cols, int e) {
    int i = blockIdx.x * blockDim.x + threadIdx.x;
    if (i < e) {
        int r = rows[i];
        int p = atomicAdd(&cursor[r], 1);
        csr_cols[p] = cols[i];
    }
}

// ------------------------- hop 0: normalize x ------------------------------
__global__ __launch_bounds__(256) void norm0_kernel(const float* __restrict__ x,
                                                    float* __restrict__ out,
                                                    int n) {
    const int lane = threadIdx.x & 31;
    const int row  = blockIdx.x * (blockDim.x >> 5) + (threadIdx.x >> 5);
    if (row >= n) return;                      // wave-uniform guard
    const size_t base = (size_t)row * HOP_D + (size_t)lane * 4;
    float4 v = *(const float4*)&x[base];
    float s = v.x * v.x + v.y * v.y + v.z * v.z + v.w * v.w;
    float total = wave_sum_wmma(s);
    // 1/max(sqrt(t), 1e-12) == rsqrt(t) for t >= 1e-24: single v_rsq_f32
    float scale = __builtin_amdgcn_rsqf(fmaxf(total, 1e-24f));
    float4 o = { v.x * scale, v.y * scale, v.z * scale, v.w * scale };
    *(float4*)&out[base] = o;
}

// ------------- fused hop: CSR gather-sum + noise + normalize ---------------
__global__ __launch_bounds__(256) void hop_kernel(const float* __restrict__ hin,
                                                  float* __restrict__ hout,
                                                  const int* __restrict__ row_ptr,
                                                  const int* __restrict__ csr_cols,
                                                  int n, int hop) {
    const int lane = threadIdx.x & 31;
    const int row  = blockIdx.x * (blockDim.x >> 5) + (threadIdx.x >> 5);
    if (row >= n) return;                      // wave-uniform guard
    const int start = row_ptr[row];
    const int end   = row_ptr[row + 1];
    const size_t loff = (size_t)lane * 4;

    float4 acc = { 0.0f, 0.0f, 0.0f, 0.0f };
    for (int b = start; b < end; b += 32) {
        const int valid = min(32, end - b);
        int col = 0;
        if (lane < valid) {
            col = csr_cols[b + lane];
            // pull the gather rows toward L2/L0 ahead of use (global_prefetch_b8)
            __builtin_prefetch(&hin[(size_t)col * HOP_D], 0, 1);
        }
        int t = 0;
        // 4-deep software pipeline: 4 independent b128 gathers in flight,
        // combined pairwise before touching the accumulator chain.
        for (; t + 4 <= valid; t += 4) {
            const int c0 = __builtin_amdgcn_readlane(col, t + 0);
            const int c1 = __builtin_amdgcn_readlane(col, t + 1);
            const int c2 = __builtin_amdgcn_readlane(col, t + 2);
            const int c3 = __builtin_amdgcn_readlane(col, t + 3);
            const float4 v0 = *(const float4*)&hin[(size_t)c0 * HOP_D + loff];
            const float4 v1 = *(const float4*)&hin[(size_t)c1 * HOP_D + loff];
            const float4 v2 = *(const float4*)&hin[(size_t)c2 * HOP_D + loff];
            const float4 v3 = *(const float4*)&hin[(size_t)c3 * HOP_D + loff];
            float4 p, q;
            p.x = v0.x + v1.x; p.y = v0.y + v1.y; p.z = v0.z + v1.z; p.w = v0.w + v1.w;
            q.x = v2.x + v3.x; q.y = v2.y + v3.y; q.z = v2.z + v3.z; q.w = v2.w + v3.w;
            acc.x += p.x + q.x; acc.y += p.y + q.y;
            acc.z += p.z + q.z; acc.w += p.w + q.w;
        }
        for (; t < valid; ++t) {
            const int c = __builtin_amdgcn_readlane(col, t);
            const float4 v = *(const float4*)&hin[(size_t)c * HOP_D + loff];
            acc.x += v.x; acc.y += v.y; acc.z += v.z; acc.w += v.w;
        }
    }

    // deterministic gaussian noise, scale 0.1, per (hop,row,feature)
    float g[4];
    unsigned seed = ((unsigned)(hop + 1) * 0x9E3779B9u) ^
                    ((unsigned)row * 0x85EBCA6Bu) ^
                    ((unsigned)lane * 0xC2B2AE35u);
    gauss4(seed, g);
    acc.x += 0.1f * g[0]; acc.y += 0.1f * g[1];
    acc.z += 0.1f * g[2]; acc.w += 0.1f * g[3];

    // row L2 norm via WMMA-assisted reduction, then scale & store
    float s = acc.x * acc.x + acc.y * acc.y + acc.z * acc.z + acc.w * acc.w;
    float total = wave_sum_wmma(s);
    float scale = __builtin_amdgcn_rsqf(fmaxf(total, 1e-24f));
    float4 o = { acc.x * scale, acc.y * scale, acc.z * scale, acc.w * scale };
    *(float4*)&hout[(size_t)row * HOP_D + loff] = o;
}

// ------------------------------ launcher -----------------------------------
extern "C" void kernel_launch(void* const* d_in, const int* in_sizes, int n_in,
                              void* d_out, int out_size, void* d_ws, size_t ws_size,
                              hipStream_t stream) {
    const float* x  = (const float*)d_in[0];
    const int*   ei = (const int*)d_in[1];     // [2, E] flattened: rows then cols
    const int N = in_sizes[0] / HOP_D;
    const int E = in_sizes[1] / 2;
    const int* rows = ei;
    const int* cols = ei + E;
    float* out = (float*)d_out;                // [NUM_HOPS+1, N, 128]

    // workspace layout (ints): row_ptr[N+1] | cursor[N] | deg[N] | csr_cols[E]
    int* ws       = (int*)d_ws;
    int* row_ptr  = ws;
    int* cursor   = ws + (N + 1);
    int* deg      = ws + (2 * N + 1);
    int* csr_cols = ws + (3 * N + 2);

    // ---- CSR build (amortized over 8 hops; removes all hop-loop atomics) ----
    hipMemsetAsync(deg, 0, sizeof(int) * (size_t)N, stream);
    degree_kernel<<<(E + 255) / 256, 256, 0, stream>>>(rows, deg, E);
    scan_kernel<<<1, 1024, 0, stream>>>(deg, row_ptr, cursor, N);
    fill_kernel<<<(E + 255) / 256, 256, 0, stream>>>(rows, cols, cursor, csr_cols, E);

    // ---- hop 0: normalize x into slab 0 ----
    const int waves_per_block = 8;             // 256 threads = 8 wave32
    const int grid = (N + waves_per_block - 1) / waves_per_block;
    norm0_kernel<<<grid, 256, 0, stream>>>(x, out, N);

    // ---- hops 1..8: fused gather + noise + normalize ----
    const size_t slab = (size_t)N * HOP_D;
    for (int k = 0; k < NUM_HOPS; ++k) {
        hop_kernel<<<grid, 256, 0, stream>>>(out + (size_t)k * slab,
                                             out + (size_t)(k + 1) * slab,
                                             row_ptr, csr_cols, N, k);
    }
}